// TopologicalDiversitySampler_18322330485331
// MI455X (gfx1250) — compile-verified
//
#include <hip/hip_runtime.h>
#include <hip/hip_bf16.h>
#include <math.h>

// ---------------------------------------------------------------------------
// TopologicalDiversitySampler for MI455X (gfx1250, wave32)
//  Phase 1: feats = features @ W^T + b via V_WMMA_F32_16X16X4_F32 with
//           LDS-staged, double-buffered W panels (bank-conflict-free) and
//           register-double-buffered A; panel loop unrolled x2 so the buffer
//           selector is compile-time (no v_movrels). Fused row L2-normalize.
//  Phase 2: 511 fused greedy iterations (distance/min_d/argmax/mask-update in
//           ONE kernel per iteration via threadfence last-block reduction).
// ---------------------------------------------------------------------------

typedef __attribute__((ext_vector_type(2))) float v2f;
typedef __attribute__((ext_vector_type(8))) float v8f;

#define ALPHA_C 0.5f
#define RDIM 128            // R (output feature dim), 8 WMMA column tiles
#define GEMM_WPB 8          // waves per block in gemm kernel
#define PK 32               // K-panel size staged in LDS
#define LDSTRIDE 36         // padded row stride (DWORDs) -> conflict-free
#define R1B 208             // blocks for attention reduction stage 1
#define SB 160              // blocks for each greedy step kernel

__device__ __forceinline__ unsigned f2ord(float f) {
  unsigned u = __float_as_uint(f);
  return (u & 0x80000000u) ? ~u : (u | 0x80000000u);
}

// --------------------------- Phase 1: WMMA GEMM -----------------------------
// One wave computes rows [m0, m0+16) x all 128 columns.
// A 16x4 f32 layout: lanes 0-15 M=lane (K=0/1 in v0/v1), lanes 16-31 K=2/3.
// B 4x16 f32 layout mirrors A with N=lane%16. B[k][n] = W[n][k] (from LDS).
__global__ void __launch_bounds__(256)
gemm_norm_kernel(const float* __restrict__ A, const float* __restrict__ W,
                 const float* __restrict__ bias, float* __restrict__ fn,
                 int Npts, int D, int nStrips) {
  __shared__ float lw[2][RDIM * LDSTRIDE];   // 2 x 18KB W panels

  const int tid   = threadIdx.x;
  const int wid   = tid >> 5;
  const int strip = blockIdx.x * GEMM_WPB + wid;
  const int lane  = tid & 31;
  const int lr    = lane & 15;
  const int half  = lane >> 4;
  const int m0    = strip * 16;

  int arow_idx = m0 + lr;
  if (arow_idx >= Npts) arow_idx = Npts - 1;          // clamp (loads only)
  const float* __restrict__ arow = A + (size_t)arow_idx * D + 2 * half;

  const int NP = D / PK;                              // # K panels

  // ---- prologue: panel 0 -> LDS buf 0, A panel 0 -> regs ----
  float4 wtmp[4];
#pragma unroll
  for (int i = 0; i < 4; ++i) {
    int c = tid + 256 * i;            // 1024 float4 chunks per panel
    int row = c >> 3;
    int kk4 = (c & 7) * 4;
    wtmp[i] = *(const float4*)(W + (size_t)row * D + kk4);
  }
#pragma unroll
  for (int i = 0; i < 4; ++i) {
    int c = tid + 256 * i;
    int row = c >> 3;
    int kk4 = (c & 7) * 4;
    *(float4*)(&lw[0][row * LDSTRIDE + kk4]) = wtmp[i];
  }
  v2f areg[2][PK / 4];
#pragma unroll
  for (int s = 0; s < PK / 4; ++s) areg[0][s] = *(const v2f*)(arow + 4 * s);
  __syncthreads();

  v8f acc[8] = {};

  // One pipeline stage with COMPILE-TIME buffer index (no movrels):
  //  - issue next panel's global loads (latency hidden behind WMMAs)
  //  - 8 k-steps x 8 column tiles of WMMA with B from LDS
  //  - store next panel to the other LDS buffer, one barrier per panel
#define GEMM_PANEL(CUR, P)                                                    \
  do {                                                                        \
    const bool more_ = ((P) + 1 < NP);                                        \
    if (more_) {                                                              \
      const float* wp_ = W + (size_t)((P) + 1) * PK;                          \
      _Pragma("unroll") for (int i = 0; i < 4; ++i) {                         \
        int c_ = tid + 256 * i;                                               \
        int row_ = c_ >> 3;                                                   \
        int kk4_ = (c_ & 7) * 4;                                              \
        wtmp[i] = *(const float4*)(wp_ + (size_t)row_ * D + kk4_);            \
      }                                                                       \
      const float* ap_ = arow + ((P) + 1) * PK;                               \
      _Pragma("unroll") for (int s = 0; s < PK / 4; ++s)                      \
          areg[(CUR) ^ 1][s] = *(const v2f*)(ap_ + 4 * s);                    \
    }                                                                         \
    _Pragma("unroll") for (int s = 0; s < PK / 4; ++s) {                      \
      v2f av_ = areg[(CUR)][s];                                               \
      _Pragma("unroll") for (int t = 0; t < 8; ++t) {                         \
        v2f bv_ = *(const v2f*)(                                              \
            &lw[(CUR)][(t * 16 + lr) * LDSTRIDE + 4 * s + 2 * half]);         \
        acc[t] = __builtin_amdgcn_wmma_f32_16x16x4_f32(                       \
            false, av_, false, bv_, (short)0, acc[t], false, false);          \
      }                                                                       \
    }                                                                         \
    if (more_) {                                                              \
      _Pragma("unroll") for (int i = 0; i < 4; ++i) {                         \
        int c_ = tid + 256 * i;                                               \
        int row_ = c_ >> 3;                                                   \
        int kk4_ = (c_ & 7) * 4;                                              \
        *(float4*)(&lw[(CUR) ^ 1][row_ * LDSTRIDE + kk4_]) = wtmp[i];         \
      }                                                                       \
    }                                                                         \
    __syncthreads();                                                          \
  } while (0)

  for (int p = 0; p + 1 < NP; p += 2) {
    GEMM_PANEL(0, p);
    GEMM_PANEL(1, p + 1);
  }
  if (NP & 1) GEMM_PANEL(0, NP - 1);    // tail (NP even for D=1024)
#undef GEMM_PANEL

  // bias: every lane's column within tile t is (t*16 + lr)
#pragma unroll
  for (int t = 0; t < 8; ++t) {
    float bv = bias[t * 16 + lr];
#pragma unroll
    for (int j = 0; j < 8; ++j) acc[t][j] += bv;
  }

  // row 1/norm: row (j + 8*half) is spread over the 16 lanes of this half
  float rinv[8];
#pragma unroll
  for (int j = 0; j < 8; ++j) {
    float s = 0.f;
#pragma unroll
    for (int t = 0; t < 8; ++t) s += acc[t][j] * acc[t][j];
    s += __shfl_xor(s, 1);
    s += __shfl_xor(s, 2);
    s += __shfl_xor(s, 4);
    s += __shfl_xor(s, 8);
    rinv[j] = 1.0f / sqrtf(s);
  }

#pragma unroll
  for (int t = 0; t < 8; ++t) {
#pragma unroll
    for (int j = 0; j < 8; ++j) {
      int row = m0 + j + 8 * half;
      if (row < Npts)
        fn[(size_t)row * RDIM + t * 16 + lr] = acc[t][j] * rinv[j];
    }
  }
}

// -------------------- attention min / max / argmax --------------------------
__global__ void __launch_bounds__(256)
attn_reduce1_kernel(const float* __restrict__ a, int n,
                    float* __restrict__ pmin, float* __restrict__ pmax,
                    unsigned long long* __restrict__ parg) {
  __shared__ float smin[256], smax[256];
  __shared__ unsigned long long sarg[256];
  float mn = __builtin_inff(), mx = -__builtin_inff();
  unsigned long long bg = 0ull;
  for (int i = blockIdx.x * 256 + threadIdx.x; i < n; i += gridDim.x * 256) {
    float v = a[i];
    mn = fminf(mn, v);
    mx = fmaxf(mx, v);
    unsigned long long pk = ((unsigned long long)f2ord(v) << 32) |
                            (unsigned long long)(0xFFFFFFFFu - (unsigned)i);
    if (pk > bg) bg = pk;
  }
  smin[threadIdx.x] = mn; smax[threadIdx.x] = mx; sarg[threadIdx.x] = bg;
  __syncthreads();
  for (int s = 128; s > 0; s >>= 1) {
    if ((int)threadIdx.x < s) {
      smin[threadIdx.x] = fminf(smin[threadIdx.x], smin[threadIdx.x + s]);
      smax[threadIdx.x] = fmaxf(smax[threadIdx.x], smax[threadIdx.x + s]);
      if (sarg[threadIdx.x + s] > sarg[threadIdx.x])
        sarg[threadIdx.x] = sarg[threadIdx.x + s];
    }
    __syncthreads();
  }
  if (threadIdx.x == 0) {
    pmin[blockIdx.x] = smin[0]; pmax[blockIdx.x] = smax[0]; parg[blockIdx.x] = sarg[0];
  }
}

__global__ void __launch_bounds__(256)
attn_reduce2_kernel(const float* __restrict__ pmin, const float* __restrict__ pmax,
                    const unsigned long long* __restrict__ parg, int nparts,
                    float* scal_f, int* scal_i, unsigned* cnt,
                    long long* __restrict__ sel) {
  __shared__ float smin[256], smax[256];
  __shared__ unsigned long long sarg[256];
  float mn = __builtin_inff(), mx = -__builtin_inff();
  unsigned long long bg = 0ull;
  for (int i = threadIdx.x; i < nparts; i += 256) {
    mn = fminf(mn, pmin[i]);
    mx = fmaxf(mx, pmax[i]);
    if (parg[i] > bg) bg = parg[i];
  }
  smin[threadIdx.x] = mn; smax[threadIdx.x] = mx; sarg[threadIdx.x] = bg;
  __syncthreads();
  for (int s = 128; s > 0; s >>= 1) {
    if ((int)threadIdx.x < s) {
      smin[threadIdx.x] = fminf(smin[threadIdx.x], smin[threadIdx.x + s]);
      smax[threadIdx.x] = fmaxf(smax[threadIdx.x], smax[threadIdx.x + s]);
      if (sarg[threadIdx.x + s] > sarg[threadIdx.x])
        sarg[threadIdx.x] = sarg[threadIdx.x + s];
    }
    __syncthreads();
  }
  if (threadIdx.x == 0) {
    unsigned first = 0xFFFFFFFFu - (unsigned)(sarg[0] & 0xFFFFFFFFull);
    scal_f[0] = smin[0];
    scal_f[1] = smax[0];
    scal_i[2] = (int)first;   // "last" selected
    *cnt = 0u;
    sel[0] = (long long)first;
  }
}

__global__ void __launch_bounds__(256)
init_kernel(const float* __restrict__ a, float* __restrict__ attn,
            float* __restrict__ mind, int* __restrict__ mask,
            const float* scal_f, const int* scal_i, int n) {
  float mn = scal_f[0], mx = scal_f[1];
  int first = scal_i[2];
  float inv = 1.0f / (mx - mn + 1e-10f);
  for (int i = blockIdx.x * 256 + threadIdx.x; i < n; i += gridDim.x * 256) {
    attn[i] = (a[i] - mn) * inv;
    mind[i] = __builtin_inff();
    mask[i] = (i == first) ? 0 : 1;
  }
}

// ---------------- Phase 2: one fused greedy iteration -----------------------
// One wave per point (2 points in flight): lane loads float4 of fn row
// (coalesced 512B), xor-reduce, lane0 updates min_d + packed argmax.
// Last block commits the selection (threadfence reduction).
__global__ void __launch_bounds__(256)
step_kernel(const float* __restrict__ fn, const float* __restrict__ attn,
            float* __restrict__ mind, int* __restrict__ mask,
            long long* __restrict__ sel, int* scal_i,
            unsigned long long* __restrict__ part, unsigned* __restrict__ cnt,
            int n, int it) {
  __shared__ __align__(16) float qs[RDIM];
  __shared__ unsigned long long wbest[8];
  const int last = scal_i[2];
  if (threadIdx.x < RDIM) qs[threadIdx.x] = fn[(size_t)last * RDIM + threadIdx.x];
  __syncthreads();

  const int wid = threadIdx.x >> 5;
  const int lane = threadIdx.x & 31;
  const float4 q = *(const float4*)(qs + lane * 4);

  unsigned long long best = 0ull;
  const int wstride = gridDim.x * 8;
  for (int p = blockIdx.x * 8 + wid; p < n; p += 2 * wstride) {
    const int p2 = p + wstride;
    const bool has2 = (p2 < n);
    const float4 v1 = *(const float4*)(fn + (size_t)p * RDIM + lane * 4);
    float a1, a2 = 0.f;
    {
      float dx = v1.x - q.x, dy = v1.y - q.y, dz = v1.z - q.z, dw = v1.w - q.w;
      a1 = dx * dx + dy * dy + dz * dz + dw * dw;
    }
    if (has2) {
      const float4 v2 = *(const float4*)(fn + (size_t)p2 * RDIM + lane * 4);
      float dx = v2.x - q.x, dy = v2.y - q.y, dz = v2.z - q.z, dw = v2.w - q.w;
      a2 = dx * dx + dy * dy + dz * dz + dw * dw;
    }
#pragma unroll
    for (int off = 16; off >= 1; off >>= 1) {
      a1 += __shfl_xor(a1, off);
      a2 += __shfl_xor(a2, off);
    }
    if (lane == 0) {
      if (mask[p]) {
        float nmd = fminf(mind[p], sqrtf(a1));
        mind[p] = nmd;
        float comb = (1.0f - ALPHA_C) * attn[p] + ALPHA_C * nmd;
        unsigned long long pk = ((unsigned long long)f2ord(comb) << 32) |
                                (unsigned long long)(0xFFFFFFFFu - (unsigned)p);
        if (pk > best) best = pk;
      }
      if (has2 && mask[p2]) {
        float nmd = fminf(mind[p2], sqrtf(a2));
        mind[p2] = nmd;
        float comb = (1.0f - ALPHA_C) * attn[p2] + ALPHA_C * nmd;
        unsigned long long pk = ((unsigned long long)f2ord(comb) << 32) |
                                (unsigned long long)(0xFFFFFFFFu - (unsigned)p2);
        if (pk > best) best = pk;
      }
    }
  }
  if (lane == 0) wbest[wid] = best;
  __syncthreads();

  if (threadIdx.x == 0) {
    unsigned long long bb = wbest[0];
    for (int w = 1; w < 8; ++w)
      if (wbest[w] > bb) bb = wbest[w];
    part[blockIdx.x] = bb;
    __threadfence();
    unsigned t = atomicAdd(cnt, 1u);
    if (t == gridDim.x - 1) {           // last block finalizes
      __threadfence();
      unsigned long long g = 0ull;
      for (int bidx = 0; bidx < (int)gridDim.x; ++bidx)
        if (part[bidx] > g) g = part[bidx];
      unsigned idx = 0xFFFFFFFFu - (unsigned)(g & 0xFFFFFFFFull);
      mask[idx]  = 0;
      sel[it]    = (long long)idx;
      scal_i[2]  = (int)idx;            // "last" for next iteration
      *cnt       = 0u;                  // self-reset for next launch
    }
  }
}

// ---------------------------------------------------------------------------
extern "C" void kernel_launch(void* const* d_in, const int* in_sizes, int n_in,
                              void* d_out, int out_size, void* d_ws, size_t ws_size,
                              hipStream_t stream) {
  const float* features  = (const float*)d_in[0];
  const float* attention = (const float*)d_in[1];
  const float* W         = (const float*)d_in[2];
  const float* bias      = (const float*)d_in[3];
  // d_in[4] is k on device; k == out_size (host-visible), avoid device reads.

  const int R    = in_sizes[3];          // 128
  const int D    = in_sizes[2] / R;      // 1024
  const int Npts = in_sizes[0] / D;      // 50000
  const int k    = out_size;             // 512
  long long* sel = (long long*)d_out;    // int64 output indices

  char* ws = (char*)d_ws;
  size_t off = 0;
  auto alloc = [&](size_t bytes, size_t align) -> char* {
    off = (off + align - 1) & ~(align - 1);
    char* p = ws + off;
    off += bytes;
    return p;
  };
  float* fn    = (float*)alloc((size_t)Npts * R * sizeof(float), 256);
  float* attn  = (float*)alloc((size_t)Npts * sizeof(float), 256);
  float* mind  = (float*)alloc((size_t)Npts * sizeof(float), 256);
  int*   mask  = (int*)alloc((size_t)Npts * sizeof(int), 256);
  unsigned long long* parg = (unsigned long long*)alloc(R1B * 8, 8);
  float* pmin  = (float*)alloc(R1B * 4, 4);
  float* pmax  = (float*)alloc(R1B * 4, 4);
  unsigned long long* part = (unsigned long long*)alloc(SB * 8, 8);
  float* scal_f = (float*)alloc(64, 64);
  int*   scal_i = (int*)scal_f;
  unsigned* cnt = (unsigned*)alloc(64, 64);
  (void)ws_size; (void)n_in;

  const int nStrips = (Npts + 15) / 16;
  const int gb = (nStrips + GEMM_WPB - 1) / GEMM_WPB;

  gemm_norm_kernel<<<gb, 256, 0, stream>>>(features, W, bias, fn, Npts, D, nStrips);
  attn_reduce1_kernel<<<R1B, 256, 0, stream>>>(attention, Npts, pmin, pmax, parg);
  attn_reduce2_kernel<<<1, 256, 0, stream>>>(pmin, pmax, parg, R1B, scal_f, scal_i, cnt, sel);
  init_kernel<<<R1B, 256, 0, stream>>>(attention, attn, mind, mask, scal_f, scal_i, Npts);
  for (int it = 1; it < k; ++it)
    step_kernel<<<SB, 256, 0, stream>>>(fn, attn, mind, mask, sel, scal_i, part, cnt, Npts, it);
}